// FourStateQuantizer_42245298323776
// MI455X (gfx1250) — compile-verified
//
#include <hip/hip_runtime.h>

// Four-state soft quantizer, fully fused elementwise map.
// out = sum_s s * softmax(-(x-s)^2 / 0.3)  over s in {-1,-0.5,0.5,1}
//
// Simplified (x^2 term cancels in softmax):
//   t   = exp(x / 0.3)            (one v_exp_f32: exp2(x * (1/0.3)*log2 e))
//   num = C1*(t^2 - t^-2) + 0.5*C2*(t - t^-1)
//   den = C1*(t^2 + t^-2) +     C2*(t + t^-1)
//   out = num / den
// with C1 = e^(-10/3), C2 = e^(-5/6).
// 3 TRANS32 ops + ~11 VALU ops per element; HBM-bound (402 MB @ 23.3 TB/s).

typedef float v4f __attribute__((ext_vector_type(4)));

__device__ __forceinline__ float fsq_one(float x) {
    const float K_EXP2 = 4.80898346962988f;   // (1/0.3) * log2(e)
    const float C1     = 0.0356739933f;        // e^(-10/3)
    const float C2     = 0.4345982085f;        // e^(-5/6)
    const float C2H    = 0.2172991043f;        // C2 / 2

    float t2 = __builtin_amdgcn_exp2f(x * K_EXP2);   // v_exp_f32 : t = e^(x/T)
    float r2 = __builtin_amdgcn_rcpf(t2);            // v_rcp_f32 : 1/t
    float t1 = t2 * t2;                              // t^2
    float r1 = r2 * r2;                              // t^-2
    float num = fmaf(C2H, t2 - r2, C1 * (t1 - r1));
    float den = fmaf(C2,  t2 + r2, C1 * (t1 + r1));
    return num * __builtin_amdgcn_rcpf(den);         // v_rcp_f32
}

__global__ __launch_bounds__(256) void FourStateQuantizer_kernel(
    const float* __restrict__ x, float* __restrict__ out, long long n) {
    const long long tid    = (long long)blockIdx.x * blockDim.x + threadIdx.x;
    const long long stride = (long long)gridDim.x * blockDim.x;
    const long long n4     = n >> 2;

    const v4f* __restrict__ x4 = reinterpret_cast<const v4f*>(x);
    v4f* __restrict__ o4       = reinterpret_cast<v4f*>(out);

    // Main loop: 128-bit non-temporal streaming (data is touched exactly once,
    // working set 402 MB > 192 MB L2 -> NT temporal hint avoids L2 thrash).
    for (long long i = tid; i < n4; i += stride) {
        v4f v = __builtin_nontemporal_load(&x4[i]);
        v4f o;
        o.x = fsq_one(v.x);
        o.y = fsq_one(v.y);
        o.z = fsq_one(v.z);
        o.w = fsq_one(v.w);
        __builtin_nontemporal_store(o, &o4[i]);
    }

    // Scalar tail (n is a multiple of 4 for this problem, but stay correct).
    for (long long i = (n4 << 2) + tid; i < n; i += stride) {
        float v = __builtin_nontemporal_load(&x[i]);
        __builtin_nontemporal_store(fsq_one(v), &out[i]);
    }
}

extern "C" void kernel_launch(void* const* d_in, const int* in_sizes, int n_in,
                              void* d_out, int out_size, void* d_ws, size_t ws_size,
                              hipStream_t stream) {
    const float* x = (const float*)d_in[0];
    float* out     = (float*)d_out;
    const long long n = (long long)in_sizes[0];   // 8388608 * 6 flat elements

    const int threads = 256;                       // 8 wave32s per block
    long long n4 = n >> 2;
    long long want = (n4 + threads - 1) / threads;
    int blocks = (int)((want < 65536LL) ? (want > 0 ? want : 1) : 65536LL);

    FourStateQuantizer_kernel<<<blocks, threads, 0, stream>>>(x, out, n);
}